// SpatialAttention_17411797418079
// MI455X (gfx1250) — compile-verified
//
#include <hip/hip_runtime.h>
#include <math.h>

// ---------------------------------------------------------------------------
// CBAM spatial attention, MI455X (gfx1250, wave32).
//   Pass 1: channel mean/max pooling  (822 MB read, 6.4 MB write)
//   Pass 2: 7x7 conv (via V_WMMA_F32_16X16X4_F32 implicit GEMM) + sigmoid +
//           broadcast multiply        (822 MB read, 822 MB NT write)
// Memory-bound: ~2.47 GB total -> ~106 us at 23.3 TB/s. All math hides under
// the memory stream, so the N-redundant WMMA conv costs nothing.
// ---------------------------------------------------------------------------

typedef float v2f __attribute__((ext_vector_type(2)));
typedef float v4f __attribute__((ext_vector_type(4)));
typedef float v8f __attribute__((ext_vector_type(8)));

#define B_  16
#define H_  224
#define W_  224
#define C_  256
#define TW  32          // pixels per block in pass 2
#define LW  (TW + 6)    // halo width: 38

// ---------------- Pass 1: per-pixel channel mean + max --------------------
// One wave32 per pixel: 256 contiguous floats, 8 per lane (2 x b128 loads).
__global__ __launch_bounds__(256) void sa_pool_kernel(
    const float* __restrict__ in, float* __restrict__ pooled) {
  const int wid  = (blockIdx.x * 256 + threadIdx.x) >> 5;   // global wave = pixel
  const int lane = threadIdx.x & 31;
  if (wid >= B_ * H_ * W_) return;

  const float* p = in + (size_t)wid * C_;
  v4f a = *(const v4f*)(p + lane * 4);          // floats [0,128)
  v4f b = *(const v4f*)(p + 128 + lane * 4);    // floats [128,256)

  float s = a.x + a.y + a.z + a.w + b.x + b.y + b.z + b.w;
  float m = fmaxf(fmaxf(fmaxf(a.x, a.y), fmaxf(a.z, a.w)),
                  fmaxf(fmaxf(b.x, b.y), fmaxf(b.z, b.w)));
#pragma unroll
  for (int off = 16; off > 0; off >>= 1) {
    s += __shfl_xor(s, off, 32);
    m  = fmaxf(m, __shfl_xor(m, off, 32));
  }
  if (lane == 0) {
    v2f r; r.x = s * (1.0f / C_); r.y = m;      // (avg, max) interleaved
    *(v2f*)(pooled + (size_t)wid * 2) = r;
  }
}

// ------- Pass 2: 7x7 conv (WMMA) + sigmoid + broadcast multiply -----------
// Block = 256 threads = 8 waves, covering 32 pixels of one image row.
// Conv as implicit GEMM: D[16 pix, 16] = A[16 pix, K=100] x B[K, 16] where
// B broadcasts the 98-tap weight vector (padded to 100) to every column.
__global__ __launch_bounds__(256) void sa_apply_kernel(
    const float* __restrict__ in, const float* __restrict__ pooled,
    const float* __restrict__ cw, const float* __restrict__ cb,
    float* __restrict__ out) {
  __shared__ float pool_s[7 * LW * 2];   // halo: rows h-3..h+3, cols w0-3..w0+34
  __shared__ float wk_s[104];            // 98 taps padded with zeros
  __shared__ float attn_s[TW];

  const int tid = threadIdx.x;
  const int w0  = blockIdx.x * TW;
  const int h   = blockIdx.y;
  const int b   = blockIdx.z;

  const size_t pixbase  = ((size_t)b * H_ + h) * W_ + w0;
  const float* src_tile = in + pixbase * C_;

  // Warm the 32 KB this block re-reads in the apply phase (global_prefetch_b8).
  __builtin_prefetch(src_tile + (size_t)tid * 32, 0, 3);

  // Stage pooled halo (zero-padded) into LDS: 7 x 38 x (avg,max).
  for (int i = tid; i < 7 * LW; i += 256) {
    int r = i / LW, c = i - r * LW;
    int gh = h + r - 3, gw = w0 + c - 3;
    v2f v = {0.0f, 0.0f};
    if (gh >= 0 && gh < H_ && gw >= 0 && gw < W_)
      v = *(const v2f*)(pooled + (((size_t)b * H_ + gh) * W_ + gw) * 2);
    pool_s[i * 2]     = v.x;
    pool_s[i * 2 + 1] = v.y;
  }
  // Stage weights, zero-padded so K=100 = 25 WMMA steps of K=4.
  for (int i = tid; i < 104; i += 256) wk_s[i] = (i < 98) ? cw[i] : 0.0f;
  __syncthreads();

  const int wave = tid >> 5;
  const int lane = tid & 31;

  if (wave < 2) {                         // wave 0 -> pixels 0-15, wave 1 -> 16-31
    const int m   = lane & 15;            // A-matrix row (output pixel in group)
    const int hi  = lane >> 4;            // half-wave selects K pair {0,1} vs {2,3}
    const int col = (wave << 4) + m;      // tile-local pixel index
    v8f acc = {};
#pragma unroll
    for (int k0 = 0; k0 < 100; k0 += 4) {
      const int k  = k0 + (hi << 1);      // this half-wave's first K
      // Patch element k -> (dy, dx, ch): k = (dy*7 + dx)*2 + ch (HWIO order).
      float ax = 0.0f, ay = 0.0f;
      if (k < 98) {
        int t = k >> 1, c = k & 1;
        int dy = t / 7, dx = t - dy * 7;
        ax = pool_s[(dy * LW + col + dx) * 2 + c];
        int k1 = k + 1;
        if (k1 < 98) {
          int t1 = k1 >> 1, c1 = k1 & 1;
          int dy1 = t1 / 7, dx1 = t1 - dy1 * 7;
          ay = pool_s[(dy1 * LW + col + dx1) * 2 + c1];
        }
      }
      v2f A  = {ax, ay};                                  // A: 16x4 f32 layout
      v2f Bv = {wk_s[k], wk_s[k + 1]};                    // B: weights bcast over N
      // D = A x B + C : v_wmma_f32_16x16x4_f32
      acc = __builtin_amdgcn_wmma_f32_16x16x4_f32(
          /*neg_a=*/false, A, /*neg_b=*/false, Bv,
          /*c_mod=*/(short)0, acc, /*reuse_a=*/false, /*reuse_b=*/false);
    }
    // D layout: VGPR r, lane-half hi -> M = r + 8*hi; all N columns identical.
    if (m == 0) {
      const float bias = cb[0];
#pragma unroll
      for (int r = 0; r < 8; ++r) {
        float x = acc[r] + bias;
        attn_s[(wave << 4) + (hi << 3) + r] = 1.0f / (1.0f + expf(-x));
      }
    }
  }
  __syncthreads();

  // Apply: wave wv handles pixels 4*wv..4*wv+3; 8 floats per lane per pixel.
#pragma unroll
  for (int pp = 0; pp < 4; ++pp) {
    const int pix = (wave << 2) + pp;
    const float a = attn_s[pix];
    const float* s = src_tile + (size_t)pix * C_;
    float*       d = out + (pixbase + pix) * C_;
    v4f x0 = *(const v4f*)(s + lane * 4);
    v4f x1 = *(const v4f*)(s + 128 + lane * 4);
    x0 *= a; x1 *= a;
    __builtin_nontemporal_store(x0, (v4f*)(d + lane * 4));        // streamed out,
    __builtin_nontemporal_store(x1, (v4f*)(d + 128 + lane * 4));  // never re-read
  }
}

// ---------------------------------------------------------------------------
extern "C" void kernel_launch(void* const* d_in, const int* in_sizes, int n_in,
                              void* d_out, int out_size, void* d_ws, size_t ws_size,
                              hipStream_t stream) {
  const float* in = (const float*)d_in[0];   // [16,224,224,256] f32
  const float* cw = (const float*)d_in[1];   // [7,7,2,1] f32 (HWIO)
  const float* cb = (const float*)d_in[2];   // [1] f32
  float* out    = (float*)d_out;             // [16,224,224,256] f32
  float* pooled = (float*)d_ws;              // [16,224,224,2] f32 scratch (6.4 MB)

  const int npix        = B_ * H_ * W_;      // 802816
  const int pool_blocks = npix / 8;          // 8 waves (pixels) per 256-thread block

  sa_pool_kernel<<<pool_blocks, 256, 0, stream>>>(in, pooled);
  sa_apply_kernel<<<dim3(W_ / TW, H_, B_), 256, 0, stream>>>(in, pooled, cw, cb, out);
}